// Embedding_51745765982547
// MI455X (gfx1250) — compile-verified
//
#include <hip/hip_runtime.h>
#include <hip/hip_bf16.h>

#define THREADS        256
#define TOK_PER_BLOCK  64
#define VOCAB          256

#if defined(__has_builtin)
#  if __has_builtin(__builtin_amdgcn_global_load_async_to_lds_b32)
#    define USE_ASYNC_LDS 1
#  else
#    define USE_ASYNC_LDS 0
#  endif
#else
#  define USE_ASYNC_LDS 0
#endif

typedef __attribute__((address_space(1))) int* as1_ip;
typedef __attribute__((address_space(3))) int* as3_ip;
typedef float v4f __attribute__((ext_vector_type(4)));

// out[t, :] = W[x[t], :]   (t = 0..ntok-1, row = dim floats)
__global__ __launch_bounds__(THREADS)
void embed_gather_kernel(const int* __restrict__ x,
                         const float* __restrict__ w,
                         float* __restrict__ out,
                         int ntok, int dim)
{
    __shared__ int s_idx[TOK_PER_BLOCK];

    const int tid         = threadIdx.x;
    const int block_start = blockIdx.x * TOK_PER_BLOCK;
    int cnt = ntok - block_start;
    if (cnt > TOK_PER_BLOCK) cnt = TOK_PER_BLOCK;
    if (cnt <= 0) return;

    // Warm the 512 KiB weight table into GL2 (emits global_prefetch_b8).
    if (blockIdx.x == 0) {
        const char* base  = (const char*)w;
        const int   bytes = VOCAB * dim * (int)sizeof(float);
        for (int off = tid * 128; off < bytes; off += THREADS * 128)
            __builtin_prefetch(base + off, 0, 1);
    }

    // Stage this block's token indices into LDS via the async-LDS path (ASYNCcnt).
#if USE_ASYNC_LDS
    if (tid < cnt) {
        int* gsrc = const_cast<int*>(x + block_start + tid);
        __builtin_amdgcn_global_load_async_to_lds_b32(
            (as1_ip)gsrc,
            (as3_ip)(&s_idx[tid]),
            /*offset=*/0, /*cpol=*/0);
    }
#  if __has_builtin(__builtin_amdgcn_s_wait_asynccnt)
    __builtin_amdgcn_s_wait_asynccnt(0);
#  else
    asm volatile("s_wait_asynccnt 0" ::: "memory");
#  endif
    __syncthreads();
#else
    if (tid < cnt) s_idx[tid] = x[block_start + tid];
    __syncthreads();
#endif

    const int wave = tid >> 5;   // 0..7
    const int lane = tid & 31;   // 0..31
    const int nq   = dim >> 2;   // float4 per row (128 for dim=512)

    // One wave per token: 32 lanes x 4 float4 = 512 floats per row.
    // Weight-row loads: regular-temporal (table stays hot in WGP$/GL2).
    // Output stores: non-temporal (128 MiB written once, never re-read).
    for (int tl = wave; tl < cnt; tl += (THREADS >> 5)) {
        int idx = s_idx[tl];                          // LDS broadcast read
        idx = __builtin_amdgcn_readfirstlane(idx);    // SGPR base for GVS addressing

        const v4f* __restrict__ wrow =
            (const v4f*)(w + (size_t)idx * (size_t)dim);
        v4f* __restrict__ orow =
            (v4f*)(out + (size_t)(block_start + tl) * (size_t)dim);

#pragma unroll 4
        for (int q = lane; q < nq; q += 32) {
            v4f v = wrow[q];                          // global_load_b128
            __builtin_nontemporal_store(v, &orow[q]); // global_store_b128 th:TH_STORE_NT
        }
    }
}

extern "C" void kernel_launch(void* const* d_in, const int* in_sizes, int n_in,
                              void* d_out, int out_size, void* d_ws, size_t ws_size,
                              hipStream_t stream)
{
    const int*   x   = (const int*)d_in[0];    // (8, 8192) int32
    const float* w   = (const float*)d_in[1];  // (256, 512) f32
    float*       out = (float*)d_out;          // (8, 8192, 512) f32

    const int ntok = in_sizes[0];              // 65536
    const int dim  = out_size / ntok;          // 512

    const int blocks = (ntok + TOK_PER_BLOCK - 1) / TOK_PER_BLOCK;
    hipLaunchKernelGGL(embed_gather_kernel, dim3(blocks), dim3(THREADS), 0, stream,
                       x, w, out, ntok, dim);
}